// UpdateWeight_39548058861894
// MI455X (gfx1250) — compile-verified
//
#include <hip/hip_runtime.h>

// ---------------------------------------------------------------------------
// UpdateWeight for MI455X (gfx1250): memory-bound, no matrix FLOPs.
//   Pass 1: global max-abs reduction of grad (int32) via async-to-LDS
//           double-buffered streaming + wave32 block reduce + atomicMax.
//   Pass 2: elementwise psto-quantize + weight update, b128 in / b128 out,
//           with global_prefetch_b8.
// Outputs (float32, concatenated): new_weight[n], grad[n], grad_exp[1].
// ---------------------------------------------------------------------------

#define TPB 256
#define RED_BLOCKS 2048

#if defined(__HIP_DEVICE_COMPILE__) && defined(__has_builtin)
#  if __has_builtin(__builtin_amdgcn_global_load_async_to_lds_b128)
#    define HAVE_ASYNC_B128 1
#  endif
#  if __has_builtin(__builtin_amdgcn_s_wait_asynccnt)
#    define HAVE_WAIT_ASYNC 1
#  endif
#endif

typedef int v4i __attribute__((ext_vector_type(4)));

#if defined(HAVE_ASYNC_B128)
__device__ __forceinline__ void async_copy16(const void* gsrc, void* ldst) {
  // (global int4* src, LDS int4* dst, imm offset, imm cpol)
  __builtin_amdgcn_global_load_async_to_lds_b128(
      (__attribute__((address_space(1))) v4i*)(gsrc),
      (__attribute__((address_space(3))) v4i*)(ldst),
      0, 0);
}
#endif

template <int N>
__device__ __forceinline__ void wait_async() {
#if defined(HAVE_WAIT_ASYNC)
  __builtin_amdgcn_s_wait_asynccnt(N);
#else
  if constexpr (N == 0) {
    asm volatile("s_wait_asynccnt 0" ::: "memory");
  } else {
    asm volatile("s_wait_asynccnt 1" ::: "memory");
  }
#endif
}

__device__ __forceinline__ unsigned uabs32(int x) {
  return x < 0 ? (0u - (unsigned)x) : (unsigned)x;
}

// Pseudo-stochastic rounding right-shift (shift >= 1), clipped to [-127,127].
__device__ __forceinline__ int psto_shift_i(int x, int shift) {
  const int rt   = x >> shift;                    // floor division by 2^shift
  const int prob = x - (rt << shift);             // remainder in [0, 2^shift)
  const int hs   = shift >> 1;
  const int qpr  = prob >> hs;
  const int prn  = (prob - (qpr << hs)) << (shift & 1);
  const int sgn  = (x > 0) - (x < 0);
  const int dec  = (qpr <= prn) ? 0 : sgn;
  int r = rt + dec;
  r = r > 127 ? 127 : (r < -127 ? -127 : r);
  return r;
}

__global__ void init_ws_kernel(unsigned* __restrict__ ws) {
  if (threadIdx.x == 0 && blockIdx.x == 0) ws[0] = 0u;
}

__global__ __launch_bounds__(TPB) void reduce_maxabs_kernel(
    const int* __restrict__ grad, long long n4, unsigned* __restrict__ ws) {
  __shared__ int4     stage[2][TPB];   // async-to-LDS double buffer (8 KB)
  __shared__ unsigned red[TPB];

  const int       tid    = threadIdx.x;
  const int4*     g4     = (const int4*)grad;
  const long long stride = (long long)gridDim.x * TPB;
  long long       i      = (long long)blockIdx.x * TPB + tid;
  unsigned        m      = 0u;

#if defined(HAVE_ASYNC_B128)
  int  cur     = 0;
  bool pending = (i < n4);
  if (pending) async_copy16(g4 + i, &stage[cur][tid]);
  while (pending) {
    const long long nxt  = i + stride;
    const bool      more = (nxt < n4);
    if (more) {
      async_copy16(g4 + nxt, &stage[cur ^ 1][tid]);
      wait_async<1>();                 // current buffer landed, next in flight
    } else {
      wait_async<0>();
    }
    asm volatile("" ::: "memory");
    const int4 v = stage[cur][tid];    // own wave's slot; no barrier needed
    m = max(m, uabs32(v.x));
    m = max(m, uabs32(v.y));
    m = max(m, uabs32(v.z));
    m = max(m, uabs32(v.w));
    i = nxt; cur ^= 1; pending = more;
  }
#else
  for (; i < n4; i += stride) {
    const int4 v = g4[i];
    m = max(m, uabs32(v.x));
    m = max(m, uabs32(v.y));
    m = max(m, uabs32(v.z));
    m = max(m, uabs32(v.w));
  }
#endif

  red[tid] = m;
  __syncthreads();
#pragma unroll
  for (int s = TPB / 2; s > 0; s >>= 1) {
    if (tid < s) red[tid] = max(red[tid], red[tid + s]);
    __syncthreads();
  }
  if (tid == 0) atomicMax(ws, red[0]);
}

__global__ __launch_bounds__(TPB) void update_kernel(
    const int* __restrict__ weight, const int* __restrict__ grad,
    const int* __restrict__ err_exp_p, const int* __restrict__ act_in_exp_p,
    const int* __restrict__ mu_p, const unsigned* __restrict__ ws,
    float* __restrict__ out, long long n) {
  const long long idx4 = (long long)blockIdx.x * TPB + threadIdx.x;
  const long long n4   = n >> 2;

  // Uniform scalars: bw = ceil(log2(maxabs)) computed bit-exactly.
  const unsigned m      = ws[0];
  const int      mu     = mu_p[0];
  const int      bw     = (m <= 1u) ? 0 : (32 - __clz((int)(m - 1u)));
  const int      shiftv = bw - mu;
  const bool     zerob  = (bw == 0);
  const bool     truncb = (shiftv < 1);

  if (idx4 == 0) {
    const int gshift = (zerob || truncb) ? 0 : shiftv;
    out[2 * n] = (float)(err_exp_p[0] + gshift + act_in_exp_p[0]);
  }
  if (idx4 >= n4) return;

  const int4* g4 = (const int4*)grad;
  const int4* w4 = (const int4*)weight;

  // CDNA5 speculative prefetch (lowers to global_prefetch_b8).
  if (idx4 + (long long)(TPB * 16) < n4) {
    __builtin_prefetch((const void*)(g4 + idx4 + TPB * 16), 0, 0);
    __builtin_prefetch((const void*)(w4 + idx4 + TPB * 16), 0, 0);
  }

  const int4 g = g4[idx4];
  const int4 w = w4[idx4];
  const int  ga[4] = {g.x, g.y, g.z, g.w};
  const int  wa[4] = {w.x, w.y, w.z, w.w};
  float og[4], ow[4];

#pragma unroll
  for (int k = 0; k < 4; ++k) {
    const int x = ga[k];
    int g8;
    if (zerob)       g8 = 0;
    else if (truncb) g8 = (int)(signed char)(x & 0xFF);   // int8 wrap
    else             g8 = psto_shift_i(x, shiftv);
    int nw = wa[k] - g8;                                  // int16-safe range
    nw = nw > 127 ? 127 : (nw < -127 ? -127 : nw);
    og[k] = (float)g8;
    ow[k] = (float)nw;
  }

  float4* outw = (float4*)out;
  float4* outg = (float4*)(out + n);
  outw[idx4] = make_float4(ow[0], ow[1], ow[2], ow[3]);
  outg[idx4] = make_float4(og[0], og[1], og[2], og[3]);
}

extern "C" void kernel_launch(void* const* d_in, const int* in_sizes, int n_in,
                              void* d_out, int out_size, void* d_ws, size_t ws_size,
                              hipStream_t stream) {
  (void)n_in; (void)out_size; (void)ws_size;
  const int* weight     = (const int*)d_in[0];
  const int* grad       = (const int*)d_in[1];
  const int* err_exp_p  = (const int*)d_in[2];
  const int* act_in_p   = (const int*)d_in[3];
  const int* mu_p       = (const int*)d_in[4];

  const long long n  = (long long)in_sizes[1];  // grad element count (n % 4 == 0)
  const long long n4 = n >> 2;

  unsigned* ws  = (unsigned*)d_ws;
  float*    out = (float*)d_out;

  hipLaunchKernelGGL(init_ws_kernel, dim3(1), dim3(64), 0, stream, ws);

  hipLaunchKernelGGL(reduce_maxabs_kernel, dim3(RED_BLOCKS), dim3(TPB), 0, stream,
                     grad, n4, ws);

  const long long ublocks = (n4 + TPB - 1) / TPB;
  hipLaunchKernelGGL(update_kernel, dim3((unsigned)ublocks), dim3(TPB), 0, stream,
                     weight, grad, err_exp_p, act_in_p, mu_p, ws, out, n);
}